// DynamicRouting_89060441849992
// MI455X (gfx1250) — compile-verified
//
#include <hip/hip_runtime.h>

// Problem constants (from reference setup_inputs)
#define NB    64
#define NIN   2048
#define NOUT  64
#define DD    16
#define ROWF  (NOUT*DD)          // 1024 floats per (b,i) row
#define EPSF  1e-9f

typedef float v8f __attribute__((ext_vector_type(8)));
typedef float v4f __attribute__((ext_vector_type(4)));
typedef float v2f __attribute__((ext_vector_type(2)));

__device__ __forceinline__ float dot4f(v4f a, v4f b) {
  return fmaf(a.x, b.x, fmaf(a.y, b.y, fmaf(a.z, b.z, a.w * b.w)));
}

__global__ __launch_bounds__(256) void zero_kernel(float* __restrict__ p, int n) {
  int i = blockIdx.x * blockDim.x + threadIdx.x;
  if (i < n) p[i] = 0.0f;
}

// ---------------------------------------------------------------------------
// Pass 0: s[b,n,d] = sum_i pred[b,i,n,d]   (uniform c folded in at squash)
// One wave per (b, n, 256-row i-chunk). WMMA f32 16x16x4 with A = all-ones
// computes 4-row column sums per instruction; C accumulates over the chunk.
// All-ones A makes the result invariant to the A/B K-slot lane layout.
// ---------------------------------------------------------------------------
__global__ __launch_bounds__(256) void pass0_wmma(const float* __restrict__ pred,
                                                  float* __restrict__ s) {
  const int tid  = threadIdx.x;
  const int lane = tid & 31;
  const int gw   = blockIdx.x * 8 + (tid >> 5);   // global wave id
  const int chunk = gw & 7;                       // 8 chunks of 256 i
  const int n     = (gw >> 3) & 63;
  const int b     = gw >> 9;
  const int sel   = (lane < 16) ? 0 : 2;          // half-wave K-row offset
  const int col   = lane & 15;                    // d column (B/D: N = lane%16)
  const float* base = pred + (size_t)b * (NIN * ROWF) + (size_t)n * DD + col;

  v2f ones; ones.x = 1.0f; ones.y = 1.0f;
  v8f cA = {}; v8f cB = {};
  const int i0 = chunk * 256;

  #pragma unroll 4
  for (int it = 0; it < 32; ++it) {
    const int iA = i0 + it * 8 + sel;
    const int iB = iA + 4;
    v2f bA, bB;
    bA.x = __builtin_nontemporal_load(base + (size_t)(iA    ) * ROWF);
    bA.y = __builtin_nontemporal_load(base + (size_t)(iA + 1) * ROWF);
    bB.x = __builtin_nontemporal_load(base + (size_t)(iB    ) * ROWF);
    bB.y = __builtin_nontemporal_load(base + (size_t)(iB + 1) * ROWF);
    cA = __builtin_amdgcn_wmma_f32_16x16x4_f32(false, ones, false, bA,
                                               (short)0, cA, false, false);
    cB = __builtin_amdgcn_wmma_f32_16x16x4_f32(false, ones, false, bB,
                                               (short)0, cB, false, false);
  }
  // every row of D equals the column sums; lane l (<16) holds column l.
  float tot = cA[0] + cB[0];
  if (lane < 16)
    atomicAdd(&s[(size_t)b * ROWF + n * DD + lane], tot);
}

// ---------------------------------------------------------------------------
// Fused routing pass (iterations 1 and 2).
//   logits  b[n] = v0[n]·p (+ v1[n]·p for PASS==2)   -- recomputed, not stored
//   c = softmax_n(b);   s[b,n,:] += c[n] * p[:,n,:]
// One wave per (b,i); lane owns capsules n0=2*lane, n0+1; v and the s
// accumulator live in VGPRs; 8 i's per wave; LDS tree-reduce across 8 waves.
// ---------------------------------------------------------------------------
template <int PASS>
__global__ __launch_bounds__(256) void route_pass(const float* __restrict__ pred,
                                                  const float* __restrict__ v0,
                                                  const float* __restrict__ v1,
                                                  float* __restrict__ s,
                                                  float* __restrict__ coupling) {
  __shared__ float red[8 * ROWF];                 // 32 KB
  const int tid  = threadIdx.x;
  const int wave = tid >> 5;
  const int lane = tid & 31;
  const int b     = blockIdx.x >> 5;              // 32 blocks per batch
  const int iBase = ((blockIdx.x & 31) * 8 + wave) * 8;
  const int n0    = lane * 2;
  const size_t predB = (size_t)b * (NIN * ROWF);

  v4f v0r[8], v1r[8];
  {
    const v4f* vp = (const v4f*)(v0 + (size_t)b * ROWF + n0 * DD);
    #pragma unroll
    for (int k = 0; k < 8; ++k) v0r[k] = vp[k];
  }
  if (PASS == 2) {
    const v4f* vp = (const v4f*)(v1 + (size_t)b * ROWF + n0 * DD);
    #pragma unroll
    for (int k = 0; k < 8; ++k) v1r[k] = vp[k];
  }

  v4f sa[8];
  #pragma unroll
  for (int k = 0; k < 8; ++k) { sa[k].x = 0.f; sa[k].y = 0.f; sa[k].z = 0.f; sa[k].w = 0.f; }

  for (int t = 0; t < 8; ++t) {
    const int i = iBase + t;
    const v4f* pp = (const v4f*)(pred + predB + (size_t)i * ROWF + n0 * DD);
    v4f p[8];
    #pragma unroll
    for (int k = 0; k < 8; ++k) p[k] = __builtin_nontemporal_load(pp + k);

    float b0 = 0.f, b1 = 0.f;
    #pragma unroll
    for (int k = 0; k < 4; ++k) {
      b0 += dot4f(p[k],     v0r[k]);
      b1 += dot4f(p[4 + k], v0r[4 + k]);
    }
    if (PASS == 2) {
      #pragma unroll
      for (int k = 0; k < 4; ++k) {
        b0 += dot4f(p[k],     v1r[k]);
        b1 += dot4f(p[4 + k], v1r[4 + k]);
      }
    }
    // softmax over the 64 capsules (2 per lane, 32 lanes)
    float mx = fmaxf(b0, b1);
    #pragma unroll
    for (int off = 16; off >= 1; off >>= 1) mx = fmaxf(mx, __shfl_xor(mx, off, 32));
    float e0 = __expf(b0 - mx), e1 = __expf(b1 - mx);
    float sum = e0 + e1;
    #pragma unroll
    for (int off = 16; off >= 1; off >>= 1) sum += __shfl_xor(sum, off, 32);
    float inv = 1.0f / sum;
    float c0 = e0 * inv, c1 = e1 * inv;

    if (PASS == 2) {
      v2f cc; cc.x = c0; cc.y = c1;
      __builtin_nontemporal_store(cc,
          (v2f*)(coupling + ((size_t)b * NIN + i) * NOUT + n0));
    }
    #pragma unroll
    for (int k = 0; k < 4; ++k) {
      sa[k].x = fmaf(c0, p[k].x, sa[k].x);   sa[k].y = fmaf(c0, p[k].y, sa[k].y);
      sa[k].z = fmaf(c0, p[k].z, sa[k].z);   sa[k].w = fmaf(c0, p[k].w, sa[k].w);
      sa[4+k].x = fmaf(c1, p[4+k].x, sa[4+k].x); sa[4+k].y = fmaf(c1, p[4+k].y, sa[4+k].y);
      sa[4+k].z = fmaf(c1, p[4+k].z, sa[4+k].z); sa[4+k].w = fmaf(c1, p[4+k].w, sa[4+k].w);
    }
  }

  // per-block reduction across the 8 waves, then global atomic accumulate
  float* myred = red + wave * ROWF + n0 * DD;
  #pragma unroll
  for (int k = 0; k < 8; ++k) ((v4f*)myred)[k] = sa[k];
  __syncthreads();
  for (int o = tid; o < ROWF; o += 256) {
    float acc = 0.f;
    #pragma unroll
    for (int w = 0; w < 8; ++w) acc += red[w * ROWF + o];
    atomicAdd(&s[(size_t)b * ROWF + o], acc);
  }
}

// ---------------------------------------------------------------------------
// squash: v = (s*scale) * n2/(1+n2) / sqrt(n2+eps); optionally emit pose/prob
// ---------------------------------------------------------------------------
__global__ __launch_bounds__(256) void squash_kernel(const float* __restrict__ s,
                                                     float scale,
                                                     float* __restrict__ v,
                                                     float* __restrict__ pose,
                                                     float* __restrict__ prob) {
  int idx = blockIdx.x * blockDim.x + threadIdx.x;
  if (idx >= NB * NOUT) return;
  const v4f* sp = (const v4f*)(s + (size_t)idx * DD);
  v4f x[4];
  float n2 = 0.f;
  #pragma unroll
  for (int k = 0; k < 4; ++k) {
    x[k] = sp[k];
    x[k].x *= scale; x[k].y *= scale; x[k].z *= scale; x[k].w *= scale;
    n2 += dot4f(x[k], x[k]);
  }
  float f = (n2 / (1.0f + n2)) / sqrtf(n2 + EPSF);
  v4f* vp = (v4f*)(v + (size_t)idx * DD);
  v4f* pp = pose ? (v4f*)(pose + (size_t)idx * DD) : nullptr;
  #pragma unroll
  for (int k = 0; k < 4; ++k) {
    v4f y = x[k];
    y.x *= f; y.y *= f; y.z *= f; y.w *= f;
    vp[k] = y;
    if (pp) pp[k] = y;
  }
  if (prob) prob[idx] = sqrtf(f * f * n2 + EPSF);
}

extern "C" void kernel_launch(void* const* d_in, const int* in_sizes, int n_in,
                              void* d_out, int out_size, void* d_ws, size_t ws_size,
                              hipStream_t stream) {
  (void)in_sizes; (void)n_in; (void)out_size; (void)ws_size;
  const float* pred = (const float*)d_in[0];
  float* out  = (float*)d_out;
  float* pose = out;                        // [B, NOUT, D]
  float* prob = out + NB * NOUT * DD;       // [B, NOUT]
  float* coup = prob + NB * NOUT;           // [B, NIN, NOUT]

  float* s  = (float*)d_ws;                 // [B, NOUT, D] accumulator
  float* v0 = s  + NB * NOUT * DD;
  float* v1 = v0 + NB * NOUT * DD;

  const int SN = NB * NOUT * DD;            // 65536

  // iteration 0: uniform coupling -> WMMA column-sum pass
  zero_kernel<<<256, 256, 0, stream>>>(s, SN);
  pass0_wmma<<<(NB * NOUT * 8) / 8, 256, 0, stream>>>(pred, s);
  squash_kernel<<<16, 256, 0, stream>>>(s, 1.0f / NOUT, v0, nullptr, nullptr);

  // iteration 1: b1 = v0·p
  zero_kernel<<<256, 256, 0, stream>>>(s, SN);
  route_pass<1><<<NB * 32, 256, 0, stream>>>(pred, v0, nullptr, s, nullptr);
  squash_kernel<<<16, 256, 0, stream>>>(s, 1.0f, v1, nullptr, nullptr);

  // iteration 2: b2 = v0·p + v1·p; emits coupling
  zero_kernel<<<256, 256, 0, stream>>>(s, SN);
  route_pass<2><<<NB * 32, 256, 0, stream>>>(pred, v0, v1, s, coup);
  squash_kernel<<<16, 256, 0, stream>>>(s, 1.0f, v1, pose, prob);
}